// GCNAlign_7610682048666
// MI455X (gfx1250) — compile-verified
//
#include <hip/hip_runtime.h>
#include <hip/hip_bf16.h>

typedef __attribute__((ext_vector_type(2))) float v2f;
typedef __attribute__((ext_vector_type(8))) float v8f;

#define NODES 50000
#define DIM   200
#define DIMP  208          // 13 * 16, padded column count
#define EDGES 800000

// ---------------- row L2-normalize: one wave per row ----------------
__global__ void gcn_normalize_kernel(const float* __restrict__ emb,
                                     float* __restrict__ x, int n_rows) {
    int wave   = (blockIdx.x * blockDim.x + threadIdx.x) >> 5;
    int lane   = threadIdx.x & 31;
    int nwaves = (gridDim.x * blockDim.x) >> 5;
    for (int row = wave; row < n_rows; row += nwaves) {
        const float* r = emb + (size_t)row * DIM;
        float ss = 0.f;
        for (int d = lane; d < DIM; d += 32) { float v = r[d]; ss += v * v; }
        for (int off = 16; off > 0; off >>= 1) ss += __shfl_xor(ss, off, 32);
        float scale = 1.0f / fmaxf(sqrtf(ss), 1e-12f);
        float* o = x + (size_t)row * DIM;
        for (int d = lane; d < DIM; d += 32) o[d] = r[d] * scale;
    }
}

// ------- gather x[src]*ew, scatter-add into h[tgt]: one wave per edge -------
__global__ void gcn_scatter_kernel(const float* __restrict__ x,
                                   const int* __restrict__ edges, // [2, E] flat
                                   const float* __restrict__ ew,
                                   float* __restrict__ h, int n_edges) {
    int wave   = (blockIdx.x * blockDim.x + threadIdx.x) >> 5;
    int lane   = threadIdx.x & 31;
    int nwaves = (gridDim.x * blockDim.x) >> 5;
    for (int e = wave; e < n_edges; e += nwaves) {
        int s   = edges[e];
        int t   = edges[n_edges + e];
        float w = ew[e];
        const float* xs = x + (size_t)s * DIM;
        float*       ht = h + (size_t)t * DIM;
        for (int d = lane; d < DIM; d += 32) {
            float v = xs[d] * w;
            (void)__hip_atomic_fetch_add(&ht[d], v, __ATOMIC_RELAXED,
                                         __HIP_MEMORY_SCOPE_AGENT);
        }
    }
}

// ---- one-time prep: Wt[n][k] = W[k][n] (n padded to 208 with zeros), bias padded ----
__global__ void gcn_prep_weights(const float* __restrict__ W,
                                 const float* __restrict__ b,
                                 float* __restrict__ Wt,
                                 float* __restrict__ bp) {
    int idx = blockIdx.x * blockDim.x + threadIdx.x;
    if (idx < DIMP * DIM) {
        int n = idx / DIM;
        int k = idx - n * DIM;
        Wt[idx] = (n < DIM) ? W[(size_t)k * DIM + n] : 0.f;
    }
    if (idx < DIMP) bp[idx] = (idx < DIM) ? b[idx] : 0.f;
}

// ---- OUT = relu(H @ W + b) via V_WMMA_F32_16X16X4_F32, one wave per 16x16 tile ----
// A(16x4 f32): lane<16 -> M=lane, regs {K=0,K=1}; lane>=16 -> M=lane-16, regs {K=2,K=3}
// B(4x16 f32): lane -> N=lane&15; same K split across lane halves / regs
// D(16x16 f32): VGPR v -> row v + 8*(lane>=16), col lane&15
__global__ void gcn_gemm_bias_relu_wmma(const float* __restrict__ H,
                                        const float* __restrict__ Wt, // [208][200]
                                        const float* __restrict__ bp, // [208]
                                        float* __restrict__ OUT) {
    const int NT = DIMP / 16;          // 13 column tiles
    const int MT = NODES / 16;         // 3125 row tiles (exact)
    int tile = blockIdx.x * (blockDim.x >> 5) + (threadIdx.x >> 5);
    if (tile >= MT * NT) return;       // wave-uniform exit (EXEC stays full for WMMA)
    int tm = tile / NT;
    int tn = tile - tm * NT;

    int lane = threadIdx.x & 31;
    int half = lane >> 4;              // 0: lanes 0-15, 1: lanes 16-31
    int l    = lane & 15;
    int m    = tm * 16 + l;            // A row for this lane
    int n    = tn * 16 + l;            // B/D column for this lane (< 208, always valid in Wt)

    const float* Hr = H  + (size_t)m * DIM + 2 * half;   // K = 2*half, 2*half+1
    const float* Wn = Wt + (size_t)n * DIM + 2 * half;
    v8f c = {};
    #pragma unroll 4
    for (int k0 = 0; k0 < DIM; k0 += 4) {
        v2f a = *(const v2f*)(Hr + k0);   // global_load_b64, unconditional
        v2f b = *(const v2f*)(Wn + k0);   // global_load_b64, unconditional
        c = __builtin_amdgcn_wmma_f32_16x16x4_f32(
                false, a, false, b, (short)0, c, false, false);
    }
    if (n < DIM) {
        float bv = bp[n];
        #pragma unroll
        for (int v = 0; v < 8; ++v) {
            int row   = tm * 16 + v + 8 * half;
            float val = c[v] + bv;
            OUT[(size_t)row * DIM + n] = val > 0.f ? val : 0.f;
        }
    }
}

extern "C" void kernel_launch(void* const* d_in, const int* in_sizes, int n_in,
                              void* d_out, int out_size, void* d_ws, size_t ws_size,
                              hipStream_t stream) {
    (void)in_sizes; (void)n_in; (void)out_size; (void)ws_size;

    const float* match_emb   = (const float*)d_in[0];
    const float* ref_emb     = (const float*)d_in[1];
    const float* conv_w      = (const float*)d_in[2];   // [200,200] row-major
    const float* conv_b      = (const float*)d_in[3];   // [200]
    const int*   match_edges = (const int*)d_in[4];     // [2, E]
    const int*   ref_edges   = (const int*)d_in[5];
    const float* match_ew    = (const float*)d_in[6];   // [E]
    const float* ref_ew      = (const float*)d_in[7];

    float* out  = (float*)d_out;                        // [2][N, D]
    float* xbuf = (float*)d_ws;                         // N*D floats (40 MB)
    float* hbuf = xbuf + (size_t)NODES * DIM;           // N*D floats (40 MB)
    float* wtp  = hbuf + (size_t)NODES * DIM;           // 208*200 floats
    float* bpp  = wtp  + (size_t)DIMP * DIM;            // 208 floats

    const size_t nd_bytes = (size_t)NODES * DIM * sizeof(float);
    const int    ntiles   = (NODES / 16) * (DIMP / 16); // 40625 wmma tiles

    // one-time: padded transpose of conv_w + padded bias (167 KB, L2-trivial)
    gcn_prep_weights<<<(DIMP * DIM + 255) / 256, 256, 0, stream>>>(
        conv_w, conv_b, wtp, bpp);

    for (int br = 0; br < 2; ++br) {
        const float* emb   = br ? ref_emb   : match_emb;
        const int*   edges = br ? ref_edges : match_edges;
        const float* ew    = br ? ref_ew    : match_ew;
        float*       outb  = out + (size_t)br * NODES * DIM;

        // x = normalize(emb): 6250 blocks * 8 waves = 1 wave/row
        gcn_normalize_kernel<<<6250, 256, 0, stream>>>(emb, xbuf, NODES);

        for (int layer = 0; layer < 2; ++layer) {
            hipMemsetAsync(hbuf, 0, nd_bytes, stream);
            // 25000 waves, ~32 edges each; rows are L2-resident (40 MB << 192 MB)
            gcn_scatter_kernel<<<3125, 256, 0, stream>>>(xbuf, edges, ew, hbuf, EDGES);
            float* dst = (layer == 1) ? outb : xbuf;
            gcn_gemm_bias_relu_wmma<<<(ntiles + 7) / 8, 256, 0, stream>>>(
                hbuf, wtp, bpp, dst);
        }
    }
}